// IMMloss_44152263803555
// MI455X (gfx1250) — compile-verified
//
#include <hip/hip_runtime.h>
#include <hip/hip_bf16.h>
#include <math.h>

#define EPS_IMM 0.006f

typedef __attribute__((ext_vector_type(2))) float v2f;
typedef __attribute__((ext_vector_type(8))) float v8f;

// One wave32 per group of M=16 particles, D=64 dims.
// Gram matrices via V_WMMA_F32_16X16X4_F32 (exact fp32), epilogue on VALU.
__global__ __launch_bounds__(256) void imm_loss_wmma_kernel(
    const float* __restrict__ xs,      // ys_t      [G*16*64]
    const float* __restrict__ ys,      // ys_r_stop [G*16*64]
    const float* __restrict__ wscale,  // w_scale   [G*16]
    const float* __restrict__ tweights,// time_weights [G*16]
    float* __restrict__ partial,       // [G] per-group results
    int G)
{
    const int lane = threadIdx.x & 31;
    const int wave = threadIdx.x >> 5;
    const int g = blockIdx.x * 8 + wave;
    if (g >= G) return;   // uniform per-wave; EXEC all-ones inside

    const int r  = lane & 15;   // matrix row owned by this lane pair
    const int hi = lane >> 4;   // 0 -> K%4 in {0,1}; 1 -> K%4 in {2,3}

    const float* xrow = xs + (size_t)g * (16 * 64) + (size_t)r * 64 + 2 * hi;
    const float* yrow = ys + (size_t)g * (16 * 64) + (size_t)r * 64 + 2 * hi;

    // A-operand layout for 16x16x4 f32 WMMA: lanes 0-15 hold (K=4c, 4c+1),
    // lanes 16-31 hold (K=4c+2, 4c+3) of row lane%16. B = transpose of the
    // same data uses identical register contents, so one copy serves both.
    v2f xa[16], ya[16];
    float px = 0.f, py = 0.f;
#pragma unroll
    for (int c = 0; c < 16; ++c) {
        xa[c] = *(const v2f*)(xrow + 4 * c);
        ya[c] = *(const v2f*)(yrow + 4 * c);
        px += xa[c][0] * xa[c][0] + xa[c][1] * xa[c][1];
        py += ya[c][0] * ya[c][0] + ya[c][1] * ya[c][1];
    }

    // Three Gram matrices: x.xT, y.yT, x.yT  (16x16, K=64 as 16 chunks of 4).
    v8f axx = {}, ayy = {}, axy = {};
#pragma unroll
    for (int c = 0; c < 16; ++c) {
        axx = __builtin_amdgcn_wmma_f32_16x16x4_f32(
                  false, xa[c], false, xa[c], (short)0, axx, false, false);
        ayy = __builtin_amdgcn_wmma_f32_16x16x4_f32(
                  false, ya[c], false, ya[c], (short)0, ayy, false, false);
        axy = __builtin_amdgcn_wmma_f32_16x16x4_f32(
                  false, xa[c], false, ya[c], (short)0, axy, false, false);
    }

    // Full row norms: lane l and lane l^16 together hold row l%16.
    float nx = px + __shfl_xor(px, 16, 32);   // ||x_{lane%16}||^2
    float ny = py + __shfl_xor(py, 16, 32);   // ||y_{lane%16}||^2
    float wl = wscale[(size_t)g * 16 + r];    // ws indexed by row i

    // C/D layout: VGPR v, lane l -> row i = v + 8*(l/16), col j = l%16.
    const int rbase = hi << 3;
    float nxr[8], nyr[8], wsr[8];
#pragma unroll
    for (int v = 0; v < 8; ++v) {
        nxr[v] = __shfl(nx, rbase + v, 32);
        nyr[v] = __shfl(ny, rbase + v, 32);
        wsr[v] = __shfl(wl, rbase + v, 32);
    }

    const float invD = 1.0f / 64.0f;
    float acc = 0.f;
#pragma unroll
    for (int v = 0; v < 8; ++v) {
        float dxx = sqrtf(fmaxf(nxr[v] + nx - 2.0f * axx[v], 0.0f));
        float dyy = sqrtf(fmaxf(nyr[v] + ny - 2.0f * ayy[v], 0.0f));
        float dxy = sqrtf(fmaxf(nxr[v] + ny - 2.0f * axy[v], 0.0f));
        dxx = fmaxf(dxx, EPS_IMM);
        dyy = fmaxf(dyy, EPS_IMM);
        dxy = fmaxf(dxy, EPS_IMM);
        const float w = -wsr[v] * invD;
        acc += __expf(w * dxx) + __expf(w * dyy) - 2.0f * __expf(w * dxy);
    }

    // Wave reduction over all 32 lanes (each lane summed 8 entries).
#pragma unroll
    for (int off = 16; off >= 1; off >>= 1)
        acc += __shfl_xor(acc, off, 32);

    if (lane == 0) {
        const float tw = tweights[(size_t)g * 16];   // [:,0]
        partial[g] = acc * tw * (1.0f / 256.0f);     // / (M*M)
    }
}

// Deterministic final reduction: out = sum(partial) / G.
__global__ __launch_bounds__(256) void imm_loss_reduce_kernel(
    const float* __restrict__ partial, float* __restrict__ out,
    int n, float scale)
{
    __shared__ float sm[256];
    float s = 0.f;
    for (int i = threadIdx.x; i < n; i += 256) s += partial[i];
    sm[threadIdx.x] = s;
    __syncthreads();
    for (int off = 128; off > 0; off >>= 1) {
        if ((int)threadIdx.x < off) sm[threadIdx.x] += sm[threadIdx.x + off];
        __syncthreads();
    }
    if (threadIdx.x == 0) out[0] = sm[0] * scale;
}

extern "C" void kernel_launch(void* const* d_in, const int* in_sizes, int n_in,
                              void* d_out, int out_size, void* d_ws, size_t ws_size,
                              hipStream_t stream) {
    const float* ys_t      = (const float*)d_in[0];
    const float* ys_r_stop = (const float*)d_in[1];
    const float* w_scale   = (const float*)d_in[2];
    const float* t_weights = (const float*)d_in[3];

    const int B = in_sizes[2];   // w_scale flat count == batch
    const int G = B / 16;        // groups of M=16 particles

    float* partial = (float*)d_ws;   // G floats of scratch

    const int blocks = (G + 7) / 8;  // 8 waves (groups) per 256-thread block
    imm_loss_wmma_kernel<<<blocks, 256, 0, stream>>>(
        ys_t, ys_r_stop, w_scale, t_weights, partial, G);

    imm_loss_reduce_kernel<<<1, 256, 0, stream>>>(
        partial, (float*)d_out, G, 1.0f / (float)G);
}